// QuDeepDTA_1425929142984
// MI455X (gfx1250) — compile-verified
//
#include <hip/hip_runtime.h>
#include <math.h>

typedef __attribute__((ext_vector_type(2))) float v2f;
typedef __attribute__((ext_vector_type(8))) float v8f;

#define CH 96

// Fused Conv1d(valid) -> (+bias, ReLU, global max over length) for one sample per block,
// implemented as implicit GEMM on V_WMMA_F32_16X16X4_F32.
//   A (16x4)  : weights tile from LDS, lane L: m = L%16, k = k0 + 2*(L>=16) + {0,1}
//   B (4x16)  : im2col of input from global, lane L: n = n0 + L%16, same k pair
//   D (16x16) : VGPR r, lanes 0-15 -> (M=r, N=lane), lanes 16-31 -> (M=8+r, N=lane-16)
// Weights are staged global->LDS with GLOBAL_LOAD_ASYNC_TO_LDS_B128 (ASYNCcnt).
template<int VOCAB, int LEN, int TAP, int NOUT, int KTOT, int OUTOFF>
__global__ __launch_bounds__(256)
void conv_relu_max_wmma(const float* __restrict__ xin,   // [B, VOCAB, LEN]
                        const float* __restrict__ w,     // [CH, KTOT] (== [CH,VOCAB,TAP] flat)
                        const float* __restrict__ bias,  // [CH]
                        float* __restrict__ xout)        // [B, 192]
{
    __shared__ alignas(16) float sW[CH * KTOT];
    __shared__ float sPart[8][CH];

    const int b   = blockIdx.x;
    const int tid = threadIdx.x;

    // --- Async stage weights to LDS (16B per lane per op, no VGPR round-trip). ---
    // LDS byte offset = low 32 bits of the generic pointer (ISA aperture rule).
    {
        const unsigned sw_lds   = (unsigned)(uintptr_t)(&sW[0]);
        const char*    gsrc     = (const char*)w;
        constexpr int  BYTES    = CH * KTOT * 4;     // multiple of 16
        for (int off = tid * 16; off < BYTES; off += 256 * 16) {
            unsigned    ldst = sw_lds + (unsigned)off;
            const char* gptr = gsrc + off;
            asm volatile("global_load_async_to_lds_b128 %0, %1, off"
                         :: "v"(ldst), "v"(gptr) : "memory");
        }
        asm volatile("s_wait_asynccnt 0" ::: "memory");
    }
    __syncthreads();

    const int lane    = tid & 31;
    const int wave    = tid >> 5;
    const int lm      = lane & 15;       // N within tile (B/D) and M within tile (A)
    const int halfSel = lane >> 4;       // selects K pair {k0,k0+1} vs {k0+2,k0+3}

    const float* __restrict__ px = xin + (size_t)b * (VOCAB * LEN);

    constexpr int NT = (NOUT + 15) / 16;
    constexpr int KQ = KTOT / 4;

    // Running pre-bias/pre-ReLU max: max_n relu(y+b) == relu(max_n y + b)
    float vmax[6][8];
    #pragma unroll
    for (int mt = 0; mt < 6; ++mt)
        #pragma unroll
        for (int r = 0; r < 8; ++r) vmax[mt][r] = -3.0e38f;

    for (int tile = wave; tile < NT; tile += 8) {
        const int n  = tile * 16 + lm;
        const int nc = (n < NOUT) ? n : (NOUT - 1);   // clamp address for padded lanes
        v8f acc[6] = {};
        for (int kq = 0; kq < KQ; ++kq) {
            const int k_a = kq * 4 + (halfSel << 1);  // even -> k_a,k_a+1 share vocab row
            const int va  = k_a / TAP;
            const int ta  = k_a % TAP;
            const float* psrc = px + va * LEN + nc + ta;
            v2f bf;
            bf.x = psrc[0];
            bf.y = psrc[1];
            #pragma unroll
            for (int mt = 0; mt < 6; ++mt) {
                const float2 wv = *(const float2*)(&sW[(mt * 16 + lm) * KTOT + k_a]);
                v2f af; af.x = wv.x; af.y = wv.y;
                acc[mt] = __builtin_amdgcn_wmma_f32_16x16x4_f32(
                    false, af, false, bf, (short)0, acc[mt], false, false);
            }
        }
        const bool valid = (n < NOUT);
        #pragma unroll
        for (int mt = 0; mt < 6; ++mt)
            #pragma unroll
            for (int r = 0; r < 8; ++r) {
                const float vv = valid ? acc[mt][r] : -3.0e38f;
                vmax[mt][r] = fmaxf(vmax[mt][r], vv);
            }
    }

    // Max over the 16 N-positions (lanes within each 16-lane half)
    #pragma unroll
    for (int mt = 0; mt < 6; ++mt)
        #pragma unroll
        for (int r = 0; r < 8; ++r) {
            float m = vmax[mt][r];
            #pragma unroll
            for (int off = 1; off < 16; off <<= 1)
                m = fmaxf(m, __shfl_xor(m, off, 32));
            if (lm == 0) sPart[wave][mt * 16 + 8 * halfSel + r] = m;
        }
    __syncthreads();

    if (tid < CH) {
        float m = sPart[0][tid];
        #pragma unroll
        for (int wv2 = 1; wv2 < 8; ++wv2) m = fmaxf(m, sPart[wv2][tid]);
        xout[(size_t)b * 192 + OUTOFF + tid] = fmaxf(m + bias[tid], 0.0f);
    }
}

// One wave32 per sample: Gram(6x32 -> 32x32) via shfl broadcasts, fused L2-normalize + dot.
__global__ __launch_bounds__(256)
void gram_readout(const float* __restrict__ x,      // [B, 192]
                  const float* __restrict__ w_aff,  // [1024]
                  const float* __restrict__ b_aff,  // [1]
                  float* __restrict__ out, int nB)
{
    const int lane = threadIdx.x & 31;
    const int samp = blockIdx.x * 8 + (threadIdx.x >> 5);
    if (samp >= nB) return;
    const float* xb = x + (size_t)samp * 192;
    float xv[6];
    #pragma unroll
    for (int i = 0; i < 6; ++i) xv[i] = xb[i * 32 + lane];
    float s1 = 0.f, s2 = 0.f;
    for (int j = 0; j < 32; ++j) {
        float g = 0.f;
        #pragma unroll
        for (int i = 0; i < 6; ++i) g += __shfl(xv[i], j, 32) * xv[i];
        s1 += g * g;                       // ||gram||^2 accumulation
        s2 += g * w_aff[j * 32 + lane];    // gram . w_aff
    }
    #pragma unroll
    for (int off = 16; off >= 1; off >>= 1) {
        s1 += __shfl_xor(s1, off, 32);
        s2 += __shfl_xor(s2, off, 32);
    }
    if (lane == 0) out[samp] = s2 / (sqrtf(s1) + 1e-12f) + b_aff[0];
}

extern "C" void kernel_launch(void* const* d_in, const int* in_sizes, int n_in,
                              void* d_out, int out_size, void* d_ws, size_t ws_size,
                              hipStream_t stream) {
    const float* protein = (const float*)d_in[0];  // [4096, 25, 1000]
    const float* ligand  = (const float*)d_in[1];  // [4096, 64, 100]
    const float* w_pro   = (const float*)d_in[2];  // [96, 25, 8]
    const float* b_pro   = (const float*)d_in[3];  // [96]
    const float* w_lig   = (const float*)d_in[4];  // [96, 64, 4]
    const float* b_lig   = (const float*)d_in[5];  // [96]
    const float* w_aff   = (const float*)d_in[6];  // [1024]
    const float* b_aff   = (const float*)d_in[7];  // [1]
    float* out  = (float*)d_out;
    float* xbuf = (float*)d_ws;                    // [B, 192] pooled features

    const int nB = out_size;                       // 4096

    // Ligand branch -> xbuf[:, 0:96]   (K = 64*4 = 256, N = 97)
    conv_relu_max_wmma<64, 100, 4, 97, 256, 0>
        <<<nB, 256, 0, stream>>>(ligand, w_lig, b_lig, xbuf);
    // Protein branch -> xbuf[:, 96:192] (K = 25*8 = 200, N = 993)
    conv_relu_max_wmma<25, 1000, 8, 993, 200, 96>
        <<<nB, 256, 0, stream>>>(protein, w_pro, b_pro, xbuf);
    // Gram + L2-normalize + QuLinear readout
    gram_readout<<<(nB + 7) / 8, 256, 0, stream>>>(xbuf, w_aff, b_aff, out, nB);
}